// Loss2_53996328845453
// MI455X (gfx1250) — compile-verified
//
#include <hip/hip_runtime.h>

// ---------------------------------------------------------------------------
// Fused low-light-loss kernel for MI455X (gfx1250, wave32, WMMA).
//
// Semantic einsums cast as one 16x7 GEMM over K = H*W:
//   A(16xK) = [ M(8) ; M^2(8) ]   (rows)
//   B(Kx7)  = [ R(3), R^2(3), 1 ] (cols)
// accumulated with V_WMMA_F32_16X16X4_F32 (4 pixels / instruction).
// All element transforms (squaring, constant columns) are materialized as
// LDS planes at stage time, so the MMA loop is pure ds_load_b64 + v_wmma.
// ---------------------------------------------------------------------------

typedef __attribute__((ext_vector_type(2))) float v2f;
typedef __attribute__((ext_vector_type(4))) float v4f;
typedef __attribute__((ext_vector_type(8))) float v8f;

namespace {
constexpr int Hh  = 512;
constexpr int Ww  = 512;
constexpr int Bb  = 16;
constexpr int RPW = 8;            // rows per workgroup
constexpr int NT  = 256;          // 8 waves
constexpr int LSTR = Ww + 4;      // padded LDS plane stride (floats)

// LDS plane indices
constexpr int PM   = 0;           // M   channels 0..7
constexpr int PM2  = 8;           // M^2 channels 0..7
constexpr int PR   = 16;          // R   channels 0..2
constexpr int PR2  = 19;          // R^2 channels 0..2
constexpr int PONE = 22;          // constant 1.0 plane
constexpr int PZER = 23;          // constant 0.0 plane
constexpr int NPL  = 24;

// workspace layout (floats)
constexpr int WS_TVH   = 0;
constexpr int WS_TVV   = 1;
constexpr int WS_COLOR = 2;                       // B*3 = 48
constexpr int WS_PATCH = 64;                      // B*32*32 = 16384
constexpr int WS_SEM   = WS_PATCH + Bb * 32 * 32; // B*256 = 4096
constexpr int WS_TOTAL = WS_SEM + Bb * 256;       // 20544 floats
}

__global__ __launch_bounds__(NT)
void fused_loss_main(const float* __restrict__ L, const float* __restrict__ R,
                     const float* __restrict__ I, const float* __restrict__ M,
                     float* __restrict__ ws)
{
  __shared__ float lds[NPL * LSTR];
  __shared__ float patchCols[32];
  __shared__ float semAcc[256];
  __shared__ float red[8];        // 0:tvH 1:tvV 2..4:color

  const int tid  = threadIdx.x;
  const int lane = tid & 31;
  const int wave = tid >> 5;
  const int wgPerB = Hh / RPW;                    // 64
  const int b  = blockIdx.x / wgPerB;
  const int h0 = (blockIdx.x % wgPerB) * RPW;

  if (tid < 32) patchCols[tid] = 0.f;
  semAcc[tid] = 0.f;
  if (tid < 8) red[tid] = 0.f;
  for (int i = tid; i < Ww; i += NT) {            // constant planes (set once)
    lds[PONE * LSTR + i] = 1.0f;
    lds[PZER * LSTR + i] = 0.0f;
  }
  __syncthreads();

  // ---- loop-invariant lane roles -> LDS base pointers for fragments ----
  const int half = lane >> 4;                     // 0 -> K0,K1 ; 1 -> K2,K3
  const int rrow = lane & 15;                     // A row: plane rrow is M/M^2
  const int col  = lane & 15;                     // B column
  const int bpl  = (col < 6) ? (PR + col) : ((col == 6) ? PONE : PZER);
  const float* pA = &lds[rrow * LSTR + half * 2];
  const float* pB = &lds[bpl  * LSTR + half * 2];
  const int qbase = wave * 64;                    // wave owns 64 px per row

  v8f acc = {0.f, 0.f, 0.f, 0.f, 0.f, 0.f, 0.f, 0.f};

  const int x = tid * 2;                          // pixel pair for scalar work
  float tvH = 0.f, tvV = 0.f;
  float c0 = 0.f, c1 = 0.f, c2 = 0.f;
  float p0 = 0.f, p1 = 0.f;

  const float* Lb = L + (size_t)b * Hh * Ww;
  const float* Rb = R + (size_t)b * 3 * Hh * Ww;
  const float* Ib = I + (size_t)b * 3 * Hh * Ww;
  const float* Mb = M + (size_t)b * 8 * Hh * Ww;

  for (int hr = 0; hr < RPW; ++hr) {
    const int h = h0 + hr;

    // ---------- phase 1: L (TV + patch sums), I (color sums) ----------
    {
      const v2f lv = *reinterpret_cast<const v2f*>(Lb + (size_t)h * Ww + x);
      const float l0 = lv.x, l1 = lv.y;
      tvH += fabsf(l0 - l1);
      if (x + 2 < Ww) {
        const float l2 = Lb[(size_t)h * Ww + x + 2];
        tvH += fabsf(l1 - l2);
      }
      if (hr > 0) tvV += fabsf(p0 - l0) + fabsf(p1 - l1);
      p0 = l0; p1 = l1;

      // 8 lanes == 16 px == one patch column: tree-reduce, one ds atomic
      float v = l0 + l1;
      v += __shfl_down(v, 4, 32);
      v += __shfl_down(v, 2, 32);
      v += __shfl_down(v, 1, 32);
      if ((lane & 7) == 0) atomicAdd(&patchCols[x >> 4], v);

      const v2f i0 = __builtin_nontemporal_load(
          reinterpret_cast<const v2f*>(Ib + (size_t)h * Ww + x));
      const v2f i1 = __builtin_nontemporal_load(
          reinterpret_cast<const v2f*>(Ib + (size_t)(Hh + h) * Ww + x));
      const v2f i2 = __builtin_nontemporal_load(
          reinterpret_cast<const v2f*>(Ib + (size_t)(2 * Hh + h) * Ww + x));
      c0 += i0.x + i0.y; c1 += i1.x + i1.y; c2 += i2.x + i2.y;
    }

    // ---- stage M,R row into LDS; squares materialized as extra planes ----
    for (int idx = tid; idx < 11 * (Ww / 4); idx += NT) {
      const int pl = idx >> 7;                    // source plane 0..10
      const int of = (idx & 127) * 4;
      const float* src = (pl < 8)
          ? (Mb + ((size_t)pl * Hh + h) * Ww + of)
          : (Rb + ((size_t)(pl - 8) * Hh + h) * Ww + of);
      const v4f v  = __builtin_nontemporal_load(reinterpret_cast<const v4f*>(src));
      const v4f vq = v * v;
      const int dst  = (pl < 8) ? pl       : (pl + 8);   // M -> 0..7, R -> 16..18
      const int dst2 = (pl < 8) ? (pl + 8) : (pl + 11);  // M2 -> 8..15, R2 -> 19..21
      *reinterpret_cast<v4f*>(&lds[dst  * LSTR + of]) = v;
      *reinterpret_cast<v4f*>(&lds[dst2 * LSTR + of]) = vq;
    }
    __syncthreads();

    // ---------- phase 2: pure ds_load + WMMA (no per-step VALU) ----------
    #pragma unroll
    for (int s = 0; s < 16; ++s) {
      const int q = qbase + s * 4;
      const v2f a = *reinterpret_cast<const v2f*>(pA + q);
      const v2f bb = *reinterpret_cast<const v2f*>(pB + q);
      acc = __builtin_amdgcn_wmma_f32_16x16x4_f32(false, a, false, bb,
                                                  (short)0, acc, false, false);
    }
    __syncthreads();
  }

  // vertical TV pair that straddles this WG's last row and the next row
  if (h0 + RPW < Hh) {
    const v2f lv = *reinterpret_cast<const v2f*>(Lb + (size_t)(h0 + RPW) * Ww + x);
    tvV += fabsf(p0 - lv.x) + fabsf(p1 - lv.y);
  }

  // ---- spill WMMA D fragment: VGPR j -> row (half*8 + j), col (lane&15) ----
  #pragma unroll
  for (int j = 0; j < 8; ++j) {
    const int idx = (half * 8 + j) * 16 + col;
    atomicAdd(&semAcc[idx], acc[j]);
  }

  // ---- wave-reduce TV / color, then LDS, then global ----
  float rH = tvH, rV = tvV, r0 = c0, r1 = c1, r2 = c2;
  for (int off = 16; off; off >>= 1) {
    rH += __shfl_down(rH, off, 32);
    rV += __shfl_down(rV, off, 32);
    r0 += __shfl_down(r0, off, 32);
    r1 += __shfl_down(r1, off, 32);
    r2 += __shfl_down(r2, off, 32);
  }
  if (lane == 0) {
    atomicAdd(&red[0], rH); atomicAdd(&red[1], rV);
    atomicAdd(&red[2], r0); atomicAdd(&red[3], r1); atomicAdd(&red[4], r2);
  }
  __syncthreads();

  atomicAdd(&ws[WS_SEM + b * 256 + tid], semAcc[tid]);
  if (tid < 32)
    atomicAdd(&ws[WS_PATCH + ((size_t)b * 32 + (h0 >> 4)) * 32 + tid],
              patchCols[tid]);
  if (tid == 0) {
    atomicAdd(&ws[WS_TVH], red[0]);
    atomicAdd(&ws[WS_TVV], red[1]);
    atomicAdd(&ws[WS_COLOR + b * 3 + 0], red[2]);
    atomicAdd(&ws[WS_COLOR + b * 3 + 1], red[3]);
    atomicAdd(&ws[WS_COLOR + b * 3 + 2], red[4]);
  }
}

__global__ __launch_bounds__(NT)
void fused_loss_final(const float* __restrict__ ws, float* __restrict__ out)
{
  __shared__ float red[NT];
  const int tid = threadIdx.x;
  float acc = 0.f;

  // exposure: mean over 16384 patches of (patch_mean - 0.6)^2, weight 10
  for (int i = tid; i < Bb * 32 * 32; i += NT) {
    const float d = ws[WS_PATCH + i] * (1.0f / 256.0f) - 0.6f;
    acc += (10.0f / 16384.0f) * d * d;
  }
  // color: weight 10, mean over batch
  for (int bi = tid; bi < Bb; bi += NT) {
    const float inv = 1.0f / (float)(Hh * Ww);
    const float r = ws[WS_COLOR + bi * 3 + 0] * inv;
    const float g = ws[WS_COLOR + bi * 3 + 1] * inv;
    const float b = ws[WS_COLOR + bi * 3 + 2] * inv;
    acc += (10.0f / 16.0f) *
           ((r - g) * (r - g) + (r - b) * (r - b) + (g - b) * (g - b));
  }
  // semantic: weight 50, sum over (b,k) / B
  for (int t = tid; t < Bb * 8; t += NT) {
    const int bi = t >> 3, k = t & 7;
    const float* S = ws + WS_SEM + bi * 256;
    const float n   = S[k * 16 + 6] + 1e-6f;
    const float sM2 = S[(k + 8) * 16 + 6];
    float vs = 0.f;
    #pragma unroll
    for (int c = 0; c < 3; ++c) {
      const float sRM   = S[k * 16 + c];
      const float sRM2  = S[(k + 8) * 16 + c];
      const float sR2M2 = S[(k + 8) * 16 + 3 + c];
      const float mean  = sRM / n;
      vs += sR2M2 - 2.f * mean * sRM2 + mean * mean * sM2;
    }
    acc += (50.0f / 16.0f) * (vs / n);
  }

  red[tid] = acc;
  __syncthreads();
  for (int off = NT / 2; off; off >>= 1) {
    if (tid < off) red[tid] += red[tid + off];
    __syncthreads();
  }
  if (tid == 0) {
    const float tv = ws[WS_TVV] * (1.0f / (16.0f * 511.0f * 512.0f)) +
                     ws[WS_TVH] * (1.0f / (16.0f * 512.0f * 511.0f));
    out[0] = red[0] + tv;   // LAMBDA_TV = 1
  }
}

extern "C" void kernel_launch(void* const* d_in, const int* in_sizes, int n_in,
                              void* d_out, int out_size, void* d_ws, size_t ws_size,
                              hipStream_t stream) {
  const float* L = (const float*)d_in[0];
  const float* R = (const float*)d_in[1];
  const float* I = (const float*)d_in[2];
  const float* M = (const float*)d_in[3];
  float* ws = (float*)d_ws;

  hipMemsetAsync(d_ws, 0, WS_TOTAL * sizeof(float), stream);

  const int grid = Bb * (Hh / RPW);   // 1024 workgroups
  fused_loss_main<<<grid, NT, 0, stream>>>(L, R, I, M, ws);
  fused_loss_final<<<1, NT, 0, stream>>>(ws, (float*)d_out);
}